// GAT_10376640987634
// MI455X (gfx1250) — compile-verified
//
#include <hip/hip_runtime.h>
#include <hip/hip_bf16.h>

typedef __attribute__((ext_vector_type(2))) float v2f;
typedef __attribute__((ext_vector_type(4))) float v4f;
typedef __attribute__((ext_vector_type(8))) float v8f;

#define NODES 4096
#define MASKW 128  // 4096/32 words per row
#define NTILES (NODES / 16)

// ---------------------------------------------------------------------------
// Pack adj (int32 0/1) into bitmask: maskbits[r*128 + w] bit b = adj[r][w*32+b]>0
// ---------------------------------------------------------------------------
__global__ __launch_bounds__(256) void pack_mask(const int* __restrict__ adj,
                                                 unsigned* __restrict__ maskbits) {
    int t = blockIdx.x * 256 + threadIdx.x;           // t < 4096*128
    long base = (long)t * 32;
    unsigned bits = 0;
    #pragma unroll
    for (int b = 0; b < 32; ++b)
        bits |= (adj[base + b] > 0 ? 1u : 0u) << b;
    maskbits[t] = bits;
}

// ---------------------------------------------------------------------------
// Generic batched WMMA f32 GEMM with compile-time strides.
// C[m][n] = sum_k A(m,k)*B(k,n);  A(m,k)=A[m*SAM+k*SAK], B(k,n)=B[k*SBK+n*SBN]
// One wave computes a 16x64 C tile via V_WMMA_F32_16X16X4_F32.
// mode: 0 = plain store, 1 = +bias then leaky_relu(0.1), 2 = +bias
// ---------------------------------------------------------------------------
template<int SAM, int SAK, int SBK, int SBN>
__global__ __launch_bounds__(256) void gemm_wmma(
        const float* __restrict__ A, long bA,
        const float* __restrict__ B, long bB,
        float* __restrict__ C, long sCm, long bC,
        const float* __restrict__ bias,
        int M, int N, int K, int mode)
{
    int batch = blockIdx.y;
    A += (long)batch * bA;  B += (long)batch * bB;  C += (long)batch * bC;

    int wave   = blockIdx.x * (blockDim.x >> 5) + (threadIdx.x >> 5);
    int mTiles = M >> 4;
    int mt = wave % mTiles;
    int nb = wave / mTiles;
    if (nb >= (N >> 6)) return;                    // wave-uniform exit

    int lane = threadIdx.x & 31;
    int half = lane >> 4;                          // 0: lanes 0-15, 1: lanes 16-31
    int l15  = lane & 15;
    int mrow = (mt << 4) + l15;                    // A-fragment row for this lane
    int n0   = nb << 6;

    const float* Arow = A + (long)mrow * SAM;

    v8f acc[4] = {};
    #pragma unroll 2
    for (int k0 = 0; k0 < K; k0 += 4) {
        int ka = k0 + half * 2;                    // A: VGPR0=K0/K2, VGPR1=K1/K3
        v2f a;
        if constexpr (SAK == 1) {
            a = *(const v2f*)(Arow + ka);          // one b64, 8B-aligned (ka even)
        } else {
            a.x = Arow[(long)ka * SAK];
            a.y = Arow[(long)(ka + 1) * SAK];
        }
        #pragma unroll
        for (int nt = 0; nt < 4; ++nt) {
            int ncol = n0 + (nt << 4) + l15;
            v2f b;                                 // B: VGPR0=K0/K2, VGPR1=K1/K3
            if constexpr (SBK == 1) {
                b = *(const v2f*)(B + ka + (long)ncol * SBN);
            } else {
                b.x = B[(long)ka * SBK + (long)ncol * SBN];
                b.y = B[(long)(ka + 1) * SBK + (long)ncol * SBN];
            }
            acc[nt] = __builtin_amdgcn_wmma_f32_16x16x4_f32(
                false, a, false, b, (short)0, acc[nt], false, false);
        }
    }

    #pragma unroll
    for (int nt = 0; nt < 4; ++nt) {
        int ncol = n0 + (nt << 4) + l15;
        float bv = (mode != 0) ? bias[ncol] : 0.0f;
        #pragma unroll
        for (int v = 0; v < 8; ++v) {
            int row = (mt << 4) + v + (half << 3); // C: VGPR v = rows v / v+8
            float val = acc[nt][v];
            if (mode == 1) { val += bv; val = val >= 0.0f ? val : 0.1f * val; }
            else if (mode == 2) { val += bv; }
            C[(long)row * sCm + ncol] = val;
        }
    }
}

// ---------------------------------------------------------------------------
// Repack Wh [4096][64] into WMMA-B fragment-major layout:
//   WhP[((jt*4+nt)*32+lane)*4 + kk] = float2{ Wh[jt*16+kk*4+half*2][nt*16+l15],
//                                             Wh[jt*16+kk*4+half*2+1][nt*16+l15] }
// so a lane's 4 K-slices are 32 contiguous bytes (two b128 loads per tile).
// One thread produces one float2; writes are fully coalesced.
// ---------------------------------------------------------------------------
__global__ __launch_bounds__(256) void repack_whb(const float* __restrict__ Wh,
                                                  float* __restrict__ WhP, int total)
{
    int t = blockIdx.x * 256 + threadIdx.x;       // one float2 per thread
    if (t >= total) return;
    int h    = t >> 17;                           // 131072 float2 per head
    int r    = t & 131071;
    int kk   = r & 3;
    int lane = (r >> 2) & 31;
    int nt   = (r >> 7) & 3;
    int jt   = r >> 9;
    int half = lane >> 4, l15 = lane & 15;
    int jr   = jt * 16 + kk * 4 + half * 2;
    int ncol = nt * 16 + l15;
    const float* src = Wh + (long)h * NODES * 64;
    v2f v;
    v.x = src[(long)jr * 64 + ncol];
    v.y = src[(long)(jr + 1) * 64 + ncol];
    ((v2f*)WhP)[t] = v;
}

// ---------------------------------------------------------------------------
// f1[t] = dot(Wh[t], a1[head]), f2[t] = dot(Wh[t], a2[head]); head = t/4096
// ---------------------------------------------------------------------------
__global__ __launch_bounds__(256) void fvec_kernel(
        const float* __restrict__ Wh, const float* __restrict__ a1,
        const float* __restrict__ a2, float* __restrict__ f1,
        float* __restrict__ f2, int total)
{
    int t = blockIdx.x * 256 + threadIdx.x;
    if (t >= total) return;
    int h = t >> 12;
    const float* row = Wh + (long)t * 64;
    const float* a1h = a1 + h * 64;
    const float* a2h = a2 + h * 64;
    float s1 = 0.0f, s2 = 0.0f;
    #pragma unroll 8
    for (int d = 0; d < 64; ++d) {
        float w = row[d];
        s1 += w * a1h[d];
        s2 += w * a2h[d];
    }
    f1[t] = s1;  f2[t] = s2;
}

// ---------------------------------------------------------------------------
// Per-row softmax stats over masked leaky_relu(f1[r]+f2[j]) : rowmax, 1/rowsum
// One thread per row; f2 staged in LDS; online max/sum.
// ---------------------------------------------------------------------------
__global__ __launch_bounds__(256) void rowstats_kernel(
        const float* __restrict__ f1, const float* __restrict__ f2,
        const unsigned* __restrict__ maskbits,
        float* __restrict__ rowmax, float* __restrict__ invsum, long bF)
{
    __shared__ float s_f2[NODES];
    int head = blockIdx.y;
    const float* f1h = f1 + (long)head * bF;
    const float* f2h = f2 + (long)head * bF;
    for (int i = threadIdx.x; i < NODES; i += 256) s_f2[i] = f2h[i];
    __syncthreads();

    int r = blockIdx.x * 256 + threadIdx.x;
    float f1v = f1h[r];
    float m = -3.0e38f, s = 0.0f;
    const unsigned* mrow = maskbits + (long)r * MASKW;
    for (int w = 0; w < MASKW; ++w) {
        unsigned bits = mrow[w];
        if (!bits) continue;
        #pragma unroll 4
        for (int b = 0; b < 32; ++b) {
            if ((bits >> b) & 1u) {
                float e = f1v + s_f2[w * 32 + b];
                e = e >= 0.0f ? e : 0.2f * e;
                if (e > m) { s = s * __expf(m - e) + 1.0f; m = e; }
                else       { s += __expf(e - m); }
            }
        }
    }
    rowmax[(long)head * bF + r] = m;
    invsum[(long)head * bF + r] = 1.0f / s;
}

// ---------------------------------------------------------------------------
// Load the 8 B-operand v4f fragments (4 n-tiles x 2) for tile jt.
// ---------------------------------------------------------------------------
__device__ __forceinline__ void load_frags(const v4f* __restrict__ Wp, int jt,
                                           int lane, v4f buf[8]) {
    #pragma unroll
    for (int nt = 0; nt < 4; ++nt) {
        long fi = (((long)jt * 4 + nt) * 32 + lane) * 2;   // v4f units
        buf[nt * 2]     = Wp[fi];
        buf[nt * 2 + 1] = Wp[fi + 1];
    }
}

// ---------------------------------------------------------------------------
// One 16-wide probability tile: exp/mask VALU work + 16 WMMAs against buf.
// ---------------------------------------------------------------------------
__device__ __forceinline__ void attn_tile(
        int jt, const unsigned* __restrict__ mrow, const float* __restrict__ s_f2,
        float f1m, float maxm, int half, const v4f buf[8], v8f acc[4])
{
    unsigned bits = (mrow[jt >> 1] >> ((jt & 1) * 16)) & 0xffffu;
    v2f pa[4];
    #pragma unroll
    for (int kk = 0; kk < 4; ++kk) {          // K-slices of 4 inside 16-wide tile
        int cc = kk * 4 + half * 2;           // column-in-tile for this lane half
        int c  = jt * 16 + cc;
        float e0 = f1m + s_f2[c];     e0 = e0 >= 0.0f ? e0 : 0.2f * e0;
        float e1 = f1m + s_f2[c + 1]; e1 = e1 >= 0.0f ? e1 : 0.2f * e1;
        pa[kk].x = ((bits >> cc) & 1u)       ? __expf(e0 - maxm) : 0.0f;
        pa[kk].y = ((bits >> (cc + 1)) & 1u) ? __expf(e1 - maxm) : 0.0f;
    }
    #pragma unroll
    for (int nt = 0; nt < 4; ++nt) {
        #pragma unroll
        for (int kk = 0; kk < 4; ++kk) {
            v4f f4 = buf[nt * 2 + (kk >> 1)];
            v2f b;
            b.x = f4[(kk & 1) * 2];
            b.y = f4[(kk & 1) * 2 + 1];
            acc[nt] = __builtin_amdgcn_wmma_f32_16x16x4_f32(
                false, pa[kk], false, b, (short)0, acc[nt], false, false);
        }
    }
}

// ---------------------------------------------------------------------------
// Fused masked-softmax attention * Wh (fragment-major WhP), ELU epilogue.
// One wave = 16 rows x full D=64. Ping-pong b128 fragment buffers (no register
// rotation); loads for tile jt+1 issue a full tile of VALU+WMMA early.
// dst[row][head*colPerHead + n] = elu( (sum_j P_mj * Wh[j][n]) * invsum[m] )
// ---------------------------------------------------------------------------
__global__ __launch_bounds__(256) void attn_kernel(
        const float* __restrict__ WhP, long bWp,
        const float* __restrict__ f1, const float* __restrict__ f2,
        const float* __restrict__ rowmax, const float* __restrict__ invsum,
        long bF, const unsigned* __restrict__ maskbits,
        float* __restrict__ dst, long dstStride, int colPerHead)
{
    __shared__ float s_f2[NODES];
    int head = blockIdx.y;
    const v4f*   Wp  = (const v4f*)(WhP + (long)head * bWp);
    const float* f1h = f1 + (long)head * bF;
    const float* f2h = f2 + (long)head * bF;
    const float* mxh = rowmax + (long)head * bF;
    const float* ish = invsum + (long)head * bF;

    for (int i = threadIdx.x; i < NODES; i += 256) s_f2[i] = f2h[i];
    __syncthreads();

    int chunk = blockIdx.x * (blockDim.x >> 5) + (threadIdx.x >> 5);
    int m0   = chunk << 4;
    int lane = threadIdx.x & 31;
    int half = lane >> 4;
    int l15  = lane & 15;
    int mA   = m0 + l15;                      // A-fragment row for this lane
    float f1m  = f1h[mA];
    float maxm = mxh[mA];
    const unsigned* mrow = maskbits + (long)mA * MASKW;

    v8f acc[4] = {};
    v4f bufA[8], bufB[8];
    load_frags(Wp, 0, lane, bufA);

    for (int jt = 0; jt < NTILES; jt += 2) {
        load_frags(Wp, jt + 1, lane, bufB);               // prefetch odd tile
        if (jt + 8 < NTILES)                              // gfx1250 prefetch path
            __builtin_prefetch(&Wp[(((long)(jt + 8) * 4) * 32 + lane) * 2], 0, 0);
        attn_tile(jt, mrow, s_f2, f1m, maxm, half, bufA, acc);

        if (jt + 2 < NTILES)
            load_frags(Wp, jt + 2, lane, bufA);           // prefetch next even tile
        attn_tile(jt + 1, mrow, s_f2, f1m, maxm, half, bufB, acc);
    }

    float invs[8];
    #pragma unroll
    for (int v = 0; v < 8; ++v) invs[v] = ish[m0 + v + (half << 3)];
    long colBase = (long)head * colPerHead;
    #pragma unroll
    for (int nt = 0; nt < 4; ++nt) {
        int ncol = (nt << 4) + l15;
        #pragma unroll
        for (int v = 0; v < 8; ++v) {
            int row = m0 + v + (half << 3);
            float val = acc[nt][v] * invs[v];
            val = val > 0.0f ? val : (__expf(val) - 1.0f);   // ELU
            dst[(long)row * dstStride + colBase + ncol] = val;
        }
    }
}

// ---------------------------------------------------------------------------
// log_softmax over axis=0 (nodes) of X[4096][64]; one block per column.
// ---------------------------------------------------------------------------
__global__ __launch_bounds__(256) void logsoftmax_col(const float* __restrict__ X,
                                                      float* __restrict__ Y)
{
    __shared__ float red[256];
    int c = blockIdx.x, tid = threadIdx.x;
    float m = -3.0e38f;
    for (int r = tid; r < NODES; r += 256) m = fmaxf(m, X[(long)r * 64 + c]);
    red[tid] = m; __syncthreads();
    for (int s = 128; s > 0; s >>= 1) {
        if (tid < s) red[tid] = fmaxf(red[tid], red[tid + s]);
        __syncthreads();
    }
    m = red[0]; __syncthreads();
    float sum = 0.0f;
    for (int r = tid; r < NODES; r += 256) sum += __expf(X[(long)r * 64 + c] - m);
    red[tid] = sum; __syncthreads();
    for (int s = 128; s > 0; s >>= 1) {
        if (tid < s) red[tid] += red[tid + s];
        __syncthreads();
    }
    float lse = m + __logf(red[0]);
    for (int r = tid; r < NODES; r += 256)
        Y[(long)r * 64 + c] = X[(long)r * 64 + c] - lse;
}

// ---------------------------------------------------------------------------
extern "C" void kernel_launch(void* const* d_in, const int* in_sizes, int n_in,
                              void* d_out, int out_size, void* d_ws, size_t ws_size,
                              hipStream_t stream) {
    const float* x   = (const float*)d_in[0];   // [4096,512]
    const int*   adj = (const int*)  d_in[1];   // [4096,4096]
    const float* W   = (const float*)d_in[2];   // [8,512,64]
    const float* a1  = (const float*)d_in[3];   // [8,64]
    const float* a2  = (const float*)d_in[4];   // [8,64]
    const float* Wo  = (const float*)d_in[5];   // [512,64]
    const float* ao1 = (const float*)d_in[6];   // [64]
    const float* ao2 = (const float*)d_in[7];   // [64]
    const float* W1  = (const float*)d_in[8];   // [512,4096]
    const float* b1  = (const float*)d_in[9];   // [512]
    const float* W2  = (const float*)d_in[10];  // [256,512]
    const float* b2  = (const float*)d_in[11];  // [256]
    float* out = (float*)d_out;                 // [64,256]

    char* ws = (char*)d_ws;
    size_t off = 0;
    auto alloc = [&](size_t bytes) -> void* {
        void* p = ws + off;
        off += bytes;
        off = (off + 255) & ~(size_t)255;
        return p;
    };
    unsigned* maskbits = (unsigned*)alloc((size_t)NODES * MASKW * 4);   // 2 MB
    float* Wh    = (float*)alloc((size_t)8 * NODES * 64 * 4);           // 8 MB
    float* WhP   = (float*)alloc((size_t)8 * NODES * 64 * 4);           // 8 MB
    float* f1    = (float*)alloc((size_t)8 * NODES * 4);
    float* f2    = (float*)alloc((size_t)8 * NODES * 4);
    float* rmax  = (float*)alloc((size_t)8 * NODES * 4);
    float* isum  = (float*)alloc((size_t)8 * NODES * 4);
    float* hcat  = (float*)alloc((size_t)NODES * 512 * 4);              // 8 MB
    float* Who   = (float*)alloc((size_t)NODES * 64 * 4);               // 1 MB
    float* WhPo  = (float*)alloc((size_t)NODES * 64 * 4);               // 1 MB
    float* f1o   = (float*)alloc((size_t)NODES * 4);
    float* f2o   = (float*)alloc((size_t)NODES * 4);
    float* rmaxo = (float*)alloc((size_t)NODES * 4);
    float* isumo = (float*)alloc((size_t)NODES * 4);
    float* outp  = (float*)alloc((size_t)NODES * 64 * 4);               // 1 MB
    float* lsm   = (float*)alloc((size_t)NODES * 64 * 4);               // 1 MB
    float* y1    = (float*)alloc((size_t)64 * 512 * 4);

    // 0) adj -> bitmask (read 64 MB once; 2 MB mask stays L2-resident)
    pack_mask<<<NODES * MASKW / 256, 256, 0, stream>>>(adj, maskbits);

    // 1) Wh[h] = x @ W[h]   (batched over 8 heads)  M=4096 N=64 K=512
    gemm_wmma<512, 1, 64, 1><<<dim3(32, 8), 256, 0, stream>>>(
        x, 0, W, (long)512 * 64,
        Wh, 64, (long)NODES * 64, nullptr, NODES, 64, 512, 0);

    // 1b) repack Wh into WMMA-B fragment-major layout (b128-friendly)
    repack_whb<<<8 * NODES * 64 / 2 / 256, 256, 0, stream>>>(Wh, WhP, 8 * NODES * 32);

    // 2) f1/f2 per head
    fvec_kernel<<<8 * NODES / 256, 256, 0, stream>>>(Wh, a1, a2, f1, f2, 8 * NODES);

    // 3) per-row softmax stats
    rowstats_kernel<<<dim3(16, 8), 256, 0, stream>>>(f1, f2, maskbits, rmax, isum, NODES);

    // 4) fused attention + ELU -> hcat[n][h*64+d]
    attn_kernel<<<dim3(32, 8), 256, 0, stream>>>(
        WhP, (long)NODES * 64, f1, f2, rmax, isum, NODES,
        maskbits, hcat, 512, 64);

    // 5) output layer: Who = hcat @ Wo   M=4096 N=64 K=512
    gemm_wmma<512, 1, 64, 1><<<dim3(32, 1), 256, 0, stream>>>(
        hcat, 0, Wo, 0,
        Who, 64, 0, nullptr, NODES, 64, 512, 0);

    repack_whb<<<NODES * 64 / 2 / 256, 256, 0, stream>>>(Who, WhPo, NODES * 32);
    fvec_kernel<<<NODES / 256, 256, 0, stream>>>(Who, ao1, ao2, f1o, f2o, NODES);
    rowstats_kernel<<<dim3(16, 1), 256, 0, stream>>>(f1o, f2o, maskbits, rmaxo, isumo, NODES);
    attn_kernel<<<dim3(32, 1), 256, 0, stream>>>(
        WhPo, 0, f1o, f2o, rmaxo, isumo, 0,
        maskbits, outp, 64, 0);

    // 6) log_softmax over nodes (axis=0)
    logsoftmax_col<<<64, 256, 0, stream>>>(outp, lsm);

    // 7) y1 = leaky_relu(lsm^T @ W1^T + b1, 0.1)  M=64 N=512 K=4096
    //    A(m,k) = lsm[k*64+m];  B(k,n) = W1[n*4096+k]
    gemm_wmma<1, 64, 1, 4096><<<dim3(4, 1), 256, 0, stream>>>(
        lsm, 0, W1, 0,
        y1, 512, 0, b1, 64, 512, 4096, 1);

    // 8) out = y1 @ W2^T + b2  M=64 N=256 K=512
    //    B(k,n) = W2[n*512+k]
    gemm_wmma<512, 1, 1, 512><<<dim3(2, 1), 256, 0, stream>>>(
        y1, 0, W2, 0,
        out, 256, 0, b2, 64, 256, 512, 2);
}